// DRuleLoss_23313082482925
// MI455X (gfx1250) — compile-verified
//
#include <hip/hip_runtime.h>

// ---------------------------------------------------------------------------
// DRuleLoss for MI455X (gfx1250, wave32).
//
// loss = (1/branch_count) * sum_{b,i} branches[i]*y[b,i]*(1 + max(y*rmax[i], y*rmin[i]))
// with branches/rmax/rmin/branch_count derived from per-row reductions of H.
//
// Bandwidth-bound: H (256MB, streamed once, > 192MB L2 -> non-temporal loads),
// y_pred (128MB, streamed once -> non-temporal). branches/rmax/rmin (3x32KB)
// are hot and reuse the cache. WMMA f32 16x16x4 is used as the wave-level
// sum-reduction engine (A = per-lane partials, B = ones).
// ---------------------------------------------------------------------------

typedef __attribute__((ext_vector_type(2))) float v2f;
typedef __attribute__((ext_vector_type(4))) float v4f;
typedef __attribute__((ext_vector_type(8))) float v8f;

// Wave32 sum reduction via V_WMMA_F32_16X16X4_F32.
// A layout (16x4 f32): VGPR0 lanes0-15 -> (M=lane,K=0), lanes16-31 -> (M=lane-16,K=2);
// VGPR1 -> K=1 / K=3. With a = {acc, 0}: A[m,0]=acc_m, A[m,2]=acc_{m+16}, rest 0.
// B = ones(4x16)  =>  D[m,n] = acc_m + acc_{m+16} for every n.
// D layout: lane n<16 VGPR r holds (M=r, N=n); lane n>=16 holds (M=r+8, N=n-16).
// So sum(d[0..7]) = sum over M=0..7 (lanes 0-15) or M=8..15 (lanes 16-31);
// one xor-16 shuffle completes the full 32-lane sum (uniform across the wave).
__device__ __forceinline__ float wave_sum_wmma(float acc) {
  v2f a; a.x = acc;  a.y = 0.0f;
  v2f b; b.x = 1.0f; b.y = 1.0f;
  v8f c = {};
  c = __builtin_amdgcn_wmma_f32_16x16x4_f32(
      /*neg_a=*/false, a, /*neg_b=*/false, b,
      /*c_mod=*/(short)0, c, /*reuse_a=*/false, /*reuse_b=*/false);
  float s = c[0] + c[1] + c[2] + c[3] + c[4] + c[5] + c[6] + c[7];
  s += __shfl_xor(s, 16);
  return s;
}

// Zero the two scalar accumulators at the head of the workspace.
__global__ void init_ws_kernel(float* ws) {
  if (threadIdx.x < 2) ws[threadIdx.x] = 0.0f;
}

// One wave per row of H: rowsum (via WMMA reduce), rowmax, rowmin (via shfl).
// branches[row] = min(rowsum, 1); atomicAdd branch_count.
__global__ void h_row_reduce_kernel(const float* __restrict__ H, int N,
                                    float* __restrict__ branches,
                                    float* __restrict__ rmax,
                                    float* __restrict__ rmin,
                                    float* __restrict__ count_acc) {
  const int wave = threadIdx.x >> 5;
  const int lane = threadIdx.x & 31;
  const int row  = blockIdx.x * (blockDim.x >> 5) + wave;
  if (row >= N) return;  // wave-uniform exit: EXEC stays all-ones for live waves

  const v4f* rowp = (const v4f*)(H + (size_t)row * (size_t)N);
  const int n4 = N >> 2;  // N is a multiple of 128

  float s  = 0.0f;
  float mx = -__builtin_inff();
  float mn =  __builtin_inff();
  for (int c = lane; c < n4; c += 32) {
    v4f v = __builtin_nontemporal_load(&rowp[c]);  // H streamed once: NT hint
    s  += (v.x + v.y) + (v.z + v.w);
    mx  = fmaxf(mx, fmaxf(fmaxf(v.x, v.y), fmaxf(v.z, v.w)));
    mn  = fminf(mn, fminf(fminf(v.x, v.y), fminf(v.z, v.w)));
  }

  s = wave_sum_wmma(s);  // full EXEC here
  #pragma unroll
  for (int o = 16; o > 0; o >>= 1) {
    mx = fmaxf(mx, __shfl_xor(mx, o));
    mn = fminf(mn, __shfl_xor(mn, o));
  }

  if (lane == 0) {
    float br = fminf(s, 1.0f);
    branches[row] = br;
    rmax[row]     = mx;
    rmin[row]     = mn;
    atomicAdd(count_acc, br);
  }
}

// Grid-stride over y_pred as float4; fold weights + nonlinearity elementwise,
// then WMMA wave reduce and one atomicAdd per wave.
__global__ void main_reduce_kernel(const float* __restrict__ y,
                                   const float* __restrict__ branches,
                                   const float* __restrict__ rmax,
                                   const float* __restrict__ rmin,
                                   int N, long long total4,
                                   float* __restrict__ acc) {
  const long long tid    = (long long)blockIdx.x * blockDim.x + threadIdx.x;
  const long long stride = (long long)gridDim.x * blockDim.x;
  const int n4 = N >> 2;

  const v4f* y4 = (const v4f*)y;
  const v4f* b4 = (const v4f*)branches;
  const v4f* x4 = (const v4f*)rmax;
  const v4f* m4 = (const v4f*)rmin;

  // Incremental column tracking avoids a 64-bit mod per iteration.
  int col  = (int)(tid % (long long)n4);
  const int step = (int)(stride % (long long)n4);

  float partial = 0.0f;
  for (long long q = tid; q < total4; q += stride) {
    v4f v  = __builtin_nontemporal_load(&y4[q]);  // y streamed once: NT hint
    v4f bb = b4[col];   // 32KB arrays: cache-resident, regular loads
    v4f rx = x4[col];
    v4f rn = m4[col];

    partial += bb.x * v.x * (1.0f + fmaxf(v.x * rx.x, v.x * rn.x));
    partial += bb.y * v.y * (1.0f + fmaxf(v.y * rx.y, v.y * rn.y));
    partial += bb.z * v.z * (1.0f + fmaxf(v.z * rx.z, v.z * rn.z));
    partial += bb.w * v.w * (1.0f + fmaxf(v.w * rx.w, v.w * rn.w));

    col += step;
    if (col >= n4) col -= n4;
  }

  float s = wave_sum_wmma(partial);  // loop reconverged: EXEC all-ones
  if ((threadIdx.x & 31) == 0) atomicAdd(acc, s);
}

__global__ void finalize_kernel(const float* __restrict__ ws, float* __restrict__ out) {
  if (threadIdx.x == 0) out[0] = ws[0] / ws[1];
}

extern "C" void kernel_launch(void* const* d_in, const int* in_sizes, int n_in,
                              void* d_out, int out_size, void* d_ws, size_t ws_size,
                              hipStream_t stream) {
  const float* y_pred = (const float*)d_in[0];  // [B, N] f32
  const float* H      = (const float*)d_in[1];  // [N, N] f32
  // d_in[2] (y_true, int64) is unused by the reference math.

  const int B = in_sizes[2];
  const int N = in_sizes[0] / B;

  // Workspace layout: [0]=loss accumulator, [1]=branch_count,
  // then branches[N], rmax[N], rmin[N] starting at a 64B-aligned offset.
  float* wsf      = (float*)d_ws;
  float* branches = wsf + 16;
  float* rmx      = branches + N;
  float* rmn      = rmx + N;

  init_ws_kernel<<<1, 64, 0, stream>>>(wsf);

  const int wavesPerBlock = 8;  // 256 threads = 8 wave32s
  const int hBlocks = (N + wavesPerBlock - 1) / wavesPerBlock;
  h_row_reduce_kernel<<<hBlocks, 32 * wavesPerBlock, 0, stream>>>(
      H, N, branches, rmx, rmn, wsf + 1);

  const long long total4 = ((long long)B * (long long)N) >> 2;
  main_reduce_kernel<<<2048, 256, 0, stream>>>(
      y_pred, branches, rmx, rmn, N, total4, wsf);

  finalize_kernel<<<1, 32, 0, stream>>>(wsf, (float*)d_out);
}